// MergeRefineOracle4Param_40991167873405
// MI455X (gfx1250) — compile-verified
//
#include <hip/hip_runtime.h>
#include <math.h>

#define B_ 4
#define C_ 3
#define H_ 720
#define W_ 1280
#define K_ 9
#define HW_ (H_*W_)        // 921600
#define CHW_ (C_*HW_)      // 2764800

// d_out layout (floats), outputs concatenated in reference return order:
// best_warped (B*C*H*W), best_idx (B), best_mse (B), all_warped (B*K*C*H*W),
// all_mse (B*K), flows (K*H*W*2)
#define OFF_BW    0ull
#define OFF_BIDX  11059200ull
#define OFF_BMSE  11059204ull
#define OFF_AW    11059208ull
#define OFF_AMSE  110592008ull
#define OFF_FLOWS 110592044ull

typedef __attribute__((ext_vector_type(2))) float v2f;
typedef __attribute__((ext_vector_type(8))) float v8f;

// Full-wave f32 sum via the matrix unit:
// A (16x4 f32): v0 = s, v1 = 0  ->  A[m][0]=s[m] (lanes 0-15), A[m][2]=s[m+16]
// B = ones(4x16)                ->  D[m][n] = s[m] + s[m+16] for every n
// Summing the 8 D VGPRs: lanes 0-15 hold rows 0-7 (= half the total),
// lanes 16-31 hold rows 8-15 (= the other half). Lanes 0 and 16 together
// carry the complete 32-lane sum -> both atomicAdd their half (no shuffle).
__device__ __forceinline__ float wave_halfsum_f32_wmma(float s) {
    v2f a; a.x = s;    a.y = 0.0f;
    v2f b; b.x = 1.0f; b.y = 1.0f;
    v8f c = {};
    v8f d = __builtin_amdgcn_wmma_f32_16x16x4_f32(
        /*neg_a=*/false, a, /*neg_b=*/false, b,
        /*c_mod=*/(short)0, c, /*reuse_a=*/false, /*reuse_b=*/false);
    return d[0] + d[1] + d[2] + d[3] + d[4] + d[5] + d[6] + d[7];
}

// One block: class k (blockIdx.y), 1024 contiguous pixels (blockIdx.x).
// Each thread owns 4 contiguous pixels of one row (W % 4 == 0 so a 4-aligned
// run never crosses a row). Streaming traffic is all b128.
__global__ __launch_bounds__(256) void warp_kernel(
    const float* __restrict__ pred, const float* __restrict__ gt,
    const float* __restrict__ cp, float* __restrict__ out,
    float* __restrict__ sums)
{
    const int k = blockIdx.y;
    const float axv = cp[k*4 + 0];
    const float ayv = cp[k*4 + 1];
    const float cxv = 1.0f / (1.0f + expf(-cp[k*4 + 2]));
    const float cyv = 1.0f / (1.0f + expf(-cp[k*4 + 3]));

    float* __restrict__ all_w = out + OFF_AW;
    float* __restrict__ flows = out + OFF_FLOWS;

    const float inv_wm1 = 1.0f / (float)(W_ - 1);
    const float inv_hm1 = 1.0f / (float)(H_ - 1);

    const int p0 = blockIdx.x * 1024 + threadIdx.x * 4;   // grid.x = HW_/1024
    const int y  = p0 / W_;
    const int x0 = p0 - y * W_;
    const float yn = (float)y * inv_hm1;

    // prefetch a later gt chunk (stays in-bounds: gt has B*C*HW floats)
    __builtin_prefetch(&gt[p0 + 1024], 0, 1);

    // per-pixel bilinear taps/weights for the 4 pixels of this thread
    int   off[4][4];
    float wgt[4][4];
    float dxq[4], dyq[4];

    #pragma unroll
    for (int q = 0; q < 4; ++q) {
        const int x = x0 + q;
        const float xn = (float)x * inv_wm1;
        const float decay = fminf(fmaxf(1.0f - 0.5f * (xn + yn), 0.0f), 1.0f);
        const float wx = cxv + (1.0f - cxv) * decay;
        const float wy = cyv + (1.0f - cyv) * decay;
        const float dx = axv * wx;
        const float dy = ayv * wy;
        dxq[q] = dx; dyq[q] = dy;

        const float ix = fminf(fmaxf((float)x + dx, 0.0f), (float)(W_ - 1));
        const float iy = fminf(fmaxf((float)y + dy, 0.0f), (float)(H_ - 1));
        const float x0f = floorf(ix), y0f = floorf(iy);
        const float fx = ix - x0f, fy = iy - y0f;
        const int xi0 = (int)x0f, yi0 = (int)y0f;
        const int xi1 = min(xi0 + 1, W_ - 1);
        const int yi1 = min(yi0 + 1, H_ - 1);

        off[q][0] = yi0 * W_ + xi0;
        off[q][1] = yi0 * W_ + xi1;
        off[q][2] = yi1 * W_ + xi0;
        off[q][3] = yi1 * W_ + xi1;
        wgt[q][0] = (1.0f - fx) * (1.0f - fy);
        wgt[q][1] = fx * (1.0f - fy);
        wgt[q][2] = (1.0f - fx) * fy;
        wgt[q][3] = fx * fy;
    }

    // flows: (K,H,W,2) -> 8 floats for 4 pixels, two b128 stores (16B aligned)
    {
        float4* flp = (float4*)(flows + (size_t)k * HW_ * 2 + (size_t)p0 * 2);
        float4 f01; f01.x = dxq[0]; f01.y = dyq[0]; f01.z = dxq[1]; f01.w = dyq[1];
        float4 f23; f23.x = dxq[2]; f23.y = dyq[2]; f23.z = dxq[3]; f23.w = dyq[3];
        flp[0] = f01;
        flp[1] = f23;
    }

    float acc[B_] = {0.f, 0.f, 0.f, 0.f};

    #pragma unroll
    for (int b = 0; b < B_; ++b) {
        #pragma unroll
        for (int c = 0; c < C_; ++c) {
            const float* __restrict__ img = pred + (size_t)(b * C_ + c) * HW_;
            float4 v;
            v.x = img[off[0][0]]*wgt[0][0] + img[off[0][1]]*wgt[0][1]
                + img[off[0][2]]*wgt[0][2] + img[off[0][3]]*wgt[0][3];
            v.y = img[off[1][0]]*wgt[1][0] + img[off[1][1]]*wgt[1][1]
                + img[off[1][2]]*wgt[1][2] + img[off[1][3]]*wgt[1][3];
            v.z = img[off[2][0]]*wgt[2][0] + img[off[2][1]]*wgt[2][1]
                + img[off[2][2]]*wgt[2][2] + img[off[2][3]]*wgt[2][3];
            v.w = img[off[3][0]]*wgt[3][0] + img[off[3][1]]*wgt[3][1]
                + img[off[3][2]]*wgt[3][2] + img[off[3][3]]*wgt[3][3];

            *(float4*)(all_w + (size_t)((b * K_ + k) * C_ + c) * HW_ + p0) = v;

            const float4 g = *(const float4*)(gt + (size_t)(b * C_ + c) * HW_ + p0);
            const float d0 = v.x - g.x, d1 = v.y - g.y;
            const float d2 = v.z - g.z, d3 = v.w - g.w;
            acc[b] = fmaf(d0, d0, acc[b]);
            acc[b] = fmaf(d1, d1, acc[b]);
            acc[b] = fmaf(d2, d2, acc[b]);
            acc[b] = fmaf(d3, d3, acc[b]);
        }
    }

    // matrix-unit reductions; EXEC all ones (no divergence above)
    #pragma unroll
    for (int b = 0; b < B_; ++b) {
        const float half = wave_halfsum_f32_wmma(acc[b]);
        if ((threadIdx.x & 15) == 0)             // lanes 0 and 16 of each wave
            atomicAdd(&sums[b * K_ + k], half);  // two complementary halves
    }
}

__global__ void finalize_kernel(const float* __restrict__ sums,
                                float* __restrict__ out,
                                int* __restrict__ widx)
{
    const int t = threadIdx.x;
    const float inv = 1.0f / (float)CHW_;
    if (t < B_ * K_) out[OFF_AMSE + t] = sums[t] * inv;
    if (t < B_) {
        float best = sums[t * K_];
        int bi = 0;
        #pragma unroll
        for (int k = 1; k < K_; ++k) {
            const float v = sums[t * K_ + k];
            if (v < best) { best = v; bi = k; }   // first-min tie-break, as argmin
        }
        out[OFF_BIDX + t] = (float)bi;
        out[OFF_BMSE + t] = best * inv;
        widx[t] = bi;
    }
}

// Pure copy of the winning class, b128 granularity.
__global__ __launch_bounds__(256) void select_kernel(
    const float4* __restrict__ all_w4, const int* __restrict__ widx,
    float4* __restrict__ bw4)
{
    const size_t CHW4 = CHW_ / 4;                 // 691200
    const size_t j0 = (size_t)blockIdx.x * 1024 + threadIdx.x;
    #pragma unroll
    for (int it = 0; it < 4; ++it) {
        const size_t j = j0 + (size_t)it * 256;   // exact: 2700*1024 == B*CHW/4
        const int b = (int)(j / CHW4);
        const size_t rest = j - (size_t)b * CHW4;
        const int idx = widx[b];
        bw4[j] = all_w4[(size_t)(b * K_ + idx) * CHW4 + rest];
    }
}

extern "C" void kernel_launch(void* const* d_in, const int* in_sizes, int n_in,
                              void* d_out, int out_size, void* d_ws, size_t ws_size,
                              hipStream_t stream) {
    const float* pred = (const float*)d_in[0];   // (B,C,H,W) f32
    const float* gt   = (const float*)d_in[1];   // (B,C,H,W) f32
    const float* cp   = (const float*)d_in[2];   // (K,4) f32
    float* out  = (float*)d_out;
    float* sums = (float*)d_ws;                        // 36 f32 accumulators
    int*   widx = (int*)((char*)d_ws + 256);           // 4 best indices

    hipMemsetAsync(d_ws, 0, 512, stream);              // zero accumulators every call

    dim3 grid1(HW_ / 1024, K_);                        // 900 x 9
    warp_kernel<<<grid1, 256, 0, stream>>>(pred, gt, cp, out, sums);

    finalize_kernel<<<1, 64, 0, stream>>>(sums, out, widx);

    select_kernel<<<(B_ * CHW_) / 4096, 256, 0, stream>>>(
        (const float4*)(out + OFF_AW), widx, (float4*)(out + OFF_BW));
}